// MOElayers_24876450579284
// MI455X (gfx1250) — compile-verified
//
#include <hip/hip_runtime.h>
#include <cstdint>
#include <cstddef>

#define DIMSZ 2048
#define INNER 5632
#define NEXP 8
#define TOKENS 2048
#define KC 64          // K-chunk staged in LDS
#define AS 68          // padded LDS row stride (floats): 16B-aligned, conflict-free
#define MT 32          // M-tile rows (2 WMMA sub-tiles per wave)

typedef float v2f __attribute__((ext_vector_type(2)));
typedef float v8f __attribute__((ext_vector_type(8)));

// ---------------- workspace layout (bytes) ----------------
#define O_CNT      0u                      // 8 ints
#define O_SEL      256u                    // TOKENS*2 ints
#define O_WGT      16640u                  // TOKENS*2 floats
#define O_LIST     33024u                  // NEXP*TOKENS ints
#define O_H        98560u                  // 2*TOKENS*INNER floats (92.3 MB)
#define O_CONTRIB  92373248u               // 2*TOKENS*DIMSZ floats (33.5 MB)

// ---------------- 1) router: softmax top-2 ----------------
__global__ __launch_bounds__(256)
void moe_router(const float* __restrict__ x, const float* __restrict__ gw,
                float* __restrict__ wgt, int* __restrict__ sel) {
  const int lane = threadIdx.x & 31;
  const int wv   = threadIdx.x >> 5;
  const int t    = blockIdx.x * 8 + wv;
  float acc[NEXP];
#pragma unroll
  for (int e = 0; e < NEXP; ++e) acc[e] = 0.0f;
  const float* xr = x + (size_t)t * DIMSZ;
  for (int d = lane; d < DIMSZ; d += 32) {
    const float xv = xr[d];
#pragma unroll
    for (int e = 0; e < NEXP; ++e) acc[e] += xv * gw[e * DIMSZ + d];
  }
#pragma unroll
  for (int e = 0; e < NEXP; ++e) {
#pragma unroll
    for (int off = 16; off > 0; off >>= 1)
      acc[e] += __shfl_xor(acc[e], off, 32);
  }
  if (lane == 0) {
    float mx = acc[0];
#pragma unroll
    for (int e = 1; e < NEXP; ++e) mx = fmaxf(mx, acc[e]);
    float p[NEXP];
#pragma unroll
    for (int e = 0; e < NEXP; ++e) p[e] = __expf(acc[e] - mx);
    int a0 = 0;
#pragma unroll
    for (int e = 1; e < NEXP; ++e) if (p[e] > p[a0]) a0 = e;
    int a1 = (a0 == 0) ? 1 : 0;
#pragma unroll
    for (int e = 0; e < NEXP; ++e) if (e != a0 && p[e] > p[a1]) a1 = e;
    const float s = p[a0] + p[a1];
    wgt[t * 2 + 0] = p[a0] / s;  sel[t * 2 + 0] = a0;
    wgt[t * 2 + 1] = p[a1] / s;  sel[t * 2 + 1] = a1;
  }
}

// ---------------- 2) compaction: per-expert slot lists ----------------
__global__ __launch_bounds__(256)
void moe_compact(const int* __restrict__ sel, int* __restrict__ cnt,
                 int* __restrict__ list) {
  const int t = blockIdx.x * blockDim.x + threadIdx.x;
  if (t >= TOKENS) return;
#pragma unroll
  for (int k = 0; k < 2; ++k) {
    const int e = sel[t * 2 + k];
    const int pos = atomicAdd(&cnt[e], 1);
    list[e * TOKENS + pos] = t * 2 + k;   // slot id encodes (token, k)
  }
}

// ---------------- 3) h = silu(x W1^T) * (x W3^T), gathered rows ----------------
__global__ __launch_bounds__(256)
void moe_gemm13(const float* __restrict__ x, const float* __restrict__ w1,
                const float* __restrict__ w3, const int* __restrict__ cnt,
                const int* __restrict__ list, float* __restrict__ hbuf) {
  const int e  = blockIdx.z;
  const int ne = cnt[e];
  const int m0 = blockIdx.y * MT;
  if (m0 >= ne) return;
  const int n0 = blockIdx.x * 128;

  __shared__ int slots[MT];
  __shared__ int rows[MT];
  __shared__ __align__(16) float As[MT * AS];
  __shared__ __align__(16) float B1[128 * AS];
  __shared__ __align__(16) float B3[128 * AS];

  const int tid = threadIdx.x;
  if (tid < MT) {
    const int m = m0 + tid;
    const int s = (m < ne) ? list[e * TOKENS + m] : -1;
    slots[tid] = s;
    rows[tid]  = (s >= 0) ? (s >> 1) : 0;
  }
  __syncthreads();

  const int lane = tid & 31;
  const int wv   = tid >> 5;
  const int mrow = lane & 15;
  const int koff = (lane >> 4) << 1;

  v8f c1[2] = {};
  v8f c3[2] = {};

  for (int kc = 0; kc < DIMSZ; kc += KC) {
    // stage A tile: 32 gathered rows x 64, 2 float4/thread
#pragma unroll
    for (int i = 0; i < 2; ++i) {
      const int idx = tid + i * 256;
      const int r = idx >> 4;
      const int q = (idx & 15) << 2;
      *(float4*)&As[r * AS + q] =
          *(const float4*)(x + (size_t)rows[r] * DIMSZ + kc + q);
    }
    // stage B tiles (w1, w3): 128x64 each, 8 float4/thread
#pragma unroll
    for (int i = 0; i < 8; ++i) {
      const int idx = tid + i * 256;
      const int r = idx >> 4;
      const int q = (idx & 15) << 2;
      const size_t g = ((size_t)e * INNER + n0 + r) * DIMSZ + kc + q;
      *(float4*)&B1[r * AS + q] = *(const float4*)(w1 + g);
      *(float4*)&B3[r * AS + q] = *(const float4*)(w3 + g);
    }
    __syncthreads();

    const float* a0p = &As[mrow * AS + koff];
    const float* a1p = &As[(16 + mrow) * AS + koff];
    const float* b1p = &B1[(wv * 16 + mrow) * AS + koff];
    const float* b3p = &B3[(wv * 16 + mrow) * AS + koff];
#pragma unroll
    for (int kk = 0; kk < KC; kk += 4) {
      v2f a0 = {a0p[kk], a0p[kk + 1]};
      v2f a1 = {a1p[kk], a1p[kk + 1]};
      v2f b1 = {b1p[kk], b1p[kk + 1]};
      v2f b3 = {b3p[kk], b3p[kk + 1]};
      c1[0] = __builtin_amdgcn_wmma_f32_16x16x4_f32(false, a0, false, b1,
                                                    (short)0, c1[0], false, false);
      c3[0] = __builtin_amdgcn_wmma_f32_16x16x4_f32(false, a0, false, b3,
                                                    (short)0, c3[0], false, false);
      c1[1] = __builtin_amdgcn_wmma_f32_16x16x4_f32(false, a1, false, b1,
                                                    (short)0, c1[1], false, false);
      c3[1] = __builtin_amdgcn_wmma_f32_16x16x4_f32(false, a1, false, b3,
                                                    (short)0, c3[1], false, false);
    }
    __syncthreads();
  }

  // epilogue: SiLU(c1)*c3, scatter by slot
  const int nIdx = n0 + wv * 16 + (lane & 15);
#pragma unroll
  for (int sub = 0; sub < 2; ++sub) {
#pragma unroll
    for (int j = 0; j < 8; ++j) {
      const int m = sub * 16 + j + ((lane >> 4) << 3);
      const int s = slots[m];
      if (s >= 0) {
        const float g = c1[sub][j];
        const float u = c3[sub][j];
        hbuf[(size_t)s * INNER + nIdx] = (g / (1.0f + __expf(-g))) * u;
      }
    }
  }
}

// ---------------- 4) contrib = (h W2^T) * routing_weight ----------------
__global__ __launch_bounds__(256)
void moe_gemm2(const float* __restrict__ hbuf, const float* __restrict__ w2,
               const int* __restrict__ cnt, const int* __restrict__ list,
               const float* __restrict__ wgt, float* __restrict__ contrib) {
  const int e  = blockIdx.z;
  const int ne = cnt[e];
  const int m0 = blockIdx.y * MT;
  if (m0 >= ne) return;
  const int n0 = blockIdx.x * 128;

  __shared__ int slots[MT];
  __shared__ int arow[MT];
  __shared__ float rw[MT];
  __shared__ __align__(16) float As[MT * AS];
  __shared__ __align__(16) float Bs[128 * AS];

  const int tid = threadIdx.x;
  if (tid < MT) {
    const int m = m0 + tid;
    const int s = (m < ne) ? list[e * TOKENS + m] : -1;
    slots[tid] = s;
    arow[tid]  = (s >= 0) ? s : 0;
    rw[tid]    = (s >= 0) ? wgt[s] : 0.0f;
  }
  __syncthreads();

  const int lane = tid & 31;
  const int wv   = tid >> 5;
  const int mrow = lane & 15;
  const int koff = (lane >> 4) << 1;

  v8f c[2] = {};

  for (int kc = 0; kc < INNER; kc += KC) {
#pragma unroll
    for (int i = 0; i < 2; ++i) {
      const int idx = tid + i * 256;
      const int r = idx >> 4;
      const int q = (idx & 15) << 2;
      *(float4*)&As[r * AS + q] =
          *(const float4*)(hbuf + (size_t)arow[r] * INNER + kc + q);
    }
#pragma unroll
    for (int i = 0; i < 8; ++i) {
      const int idx = tid + i * 256;
      const int r = idx >> 4;
      const int q = (idx & 15) << 2;
      *(float4*)&Bs[r * AS + q] =
          *(const float4*)(w2 + ((size_t)e * DIMSZ + n0 + r) * INNER + kc + q);
    }
    __syncthreads();

    const float* a0p = &As[mrow * AS + koff];
    const float* a1p = &As[(16 + mrow) * AS + koff];
    const float* bp  = &Bs[(wv * 16 + mrow) * AS + koff];
#pragma unroll
    for (int kk = 0; kk < KC; kk += 4) {
      v2f a0 = {a0p[kk], a0p[kk + 1]};
      v2f a1 = {a1p[kk], a1p[kk + 1]};
      v2f b  = {bp[kk],  bp[kk + 1]};
      c[0] = __builtin_amdgcn_wmma_f32_16x16x4_f32(false, a0, false, b,
                                                   (short)0, c[0], false, false);
      c[1] = __builtin_amdgcn_wmma_f32_16x16x4_f32(false, a1, false, b,
                                                   (short)0, c[1], false, false);
    }
    __syncthreads();
  }

  const int nIdx = n0 + wv * 16 + (lane & 15);
#pragma unroll
  for (int sub = 0; sub < 2; ++sub) {
#pragma unroll
    for (int j = 0; j < 8; ++j) {
      const int m = sub * 16 + j + ((lane >> 4) << 3);
      const int s = slots[m];
      if (s >= 0) contrib[(size_t)s * DIMSZ + nIdx] = c[sub][j] * rw[m];
    }
  }
}

// ---------------- 5) out[t] = contrib[2t] + contrib[2t+1] ----------------
__global__ __launch_bounds__(256)
void moe_finalize(const float* __restrict__ contrib, float* __restrict__ out) {
  const int DQ = DIMSZ / 4;
  const int i = blockIdx.x * blockDim.x + threadIdx.x;   // over TOKENS*DQ
  const int t = i / DQ;
  const int r = i - t * DQ;
  const float4* c = (const float4*)contrib;
  const float4 a = c[(size_t)(2 * t) * DQ + r];
  const float4 b = c[(size_t)(2 * t + 1) * DQ + r];
  float4 o;
  o.x = a.x + b.x; o.y = a.y + b.y; o.z = a.z + b.z; o.w = a.w + b.w;
  ((float4*)out)[(size_t)t * DQ + r] = o;
}

extern "C" void kernel_launch(void* const* d_in, const int* in_sizes, int n_in,
                              void* d_out, int out_size, void* d_ws, size_t ws_size,
                              hipStream_t stream) {
  const float* x  = (const float*)d_in[0];   // hidden_states [2,1024,2048]
  const float* gw = (const float*)d_in[1];   // gate_w [8,2048]
  const float* w1 = (const float*)d_in[2];   // [8,5632,2048]
  const float* w2 = (const float*)d_in[3];   // [8,2048,5632]
  const float* w3 = (const float*)d_in[4];   // [8,5632,2048]
  float* out = (float*)d_out;

  char* ws = (char*)d_ws;
  int*   cnt     = (int*)(ws + O_CNT);
  int*   sel     = (int*)(ws + O_SEL);
  float* wgt     = (float*)(ws + O_WGT);
  int*   list    = (int*)(ws + O_LIST);
  float* hbuf    = (float*)(ws + O_H);
  float* contrib = (float*)(ws + O_CONTRIB);

  hipMemsetAsync(cnt, 0, NEXP * sizeof(int), stream);

  moe_router<<<TOKENS / 8, 256, 0, stream>>>(x, gw, wgt, sel);
  moe_compact<<<TOKENS / 256, 256, 0, stream>>>(sel, cnt, list);

  dim3 gA(INNER / 128, TOKENS / MT, NEXP);
  moe_gemm13<<<gA, 256, 0, stream>>>(x, w1, w3, cnt, list, hbuf);

  dim3 gB(DIMSZ / 128, TOKENS / MT, NEXP);
  moe_gemm2<<<gB, 256, 0, stream>>>(hbuf, w2, cnt, list, wgt, contrib);

  moe_finalize<<<(TOKENS * (DIMSZ / 4)) / 256, 256, 0, stream>>>(contrib, out);
}